// GaussPSF_51737176048392
// MI455X (gfx1250) — compile-verified
//
#include <hip/hip_runtime.h>
#include <stdint.h>

// Problem dims (fixed by reference): image (3,1024,1024) f32, weights (1,1024,1024) f32,
// gauss_params (3) f32. Output (3,1024,1024) f32.
#define IMG_H 1024
#define IMG_W 1024
#define HW    (IMG_H * IMG_W)
#define KRAD  3              // kernel 7x7, radius 3
#define TILE  32
#define TROWS (TILE + 2 * KRAD)   // 38 halo rows
#define TPITCH 40                 // halo cols padded to 40 (origin bx*32-4, 16B aligned)
#define NPIX  (TROWS * TPITCH)    // 1520 tile pixels
#define EPS   1e-5f
#define LOG2E 1.44269504088896340736f

// ---- gfx1250 async global->LDS copy (ASYNCcnt path), guarded ----------------
#if defined(__gfx1250__) && __has_builtin(__builtin_amdgcn_global_load_async_to_lds_b32)
#define HAVE_ASYNC_LDS 1
typedef __attribute__((address_space(1))) int GInt;   // '__device__ int *'
typedef __attribute__((address_space(3))) int LInt;   // '__shared__ int *'
#endif

__device__ __forceinline__ void cp32_g2l(const float* __restrict__ g, float* l) {
#ifdef HAVE_ASYNC_LDS
    // (global src, lds dst, imm offset, imm cpol) -> global_load_async_to_lds_b32
    __builtin_amdgcn_global_load_async_to_lds_b32((GInt*)(void*)g, (LInt*)(void*)l, 0, 0);
#else
    *l = *g;   // fallback: global_load_b32 + ds_store_b32 through VGPRs
#endif
}

__device__ __forceinline__ float fast_rcp(float x) {
#if __has_builtin(__builtin_amdgcn_rcpf)
    return __builtin_amdgcn_rcpf(x);       // v_rcp_f32
#else
    return 1.0f / x;
#endif
}

__device__ __forceinline__ float fast_exp2(float x) {
#if __has_builtin(__builtin_amdgcn_exp2f)
    return __builtin_amdgcn_exp2f(x);      // v_exp_f32
#else
    return exp2f(x);
#endif
}

__global__ __launch_bounds__(256) void gauss_psf_kernel(
    const float* __restrict__ img,   // (3, H, W)
    const float* __restrict__ wts,   // (H, W) broadcast over channels
    const float* __restrict__ gp,    // (sx, sy, rho)
    float* __restrict__ out)         // (3, H, W)
{
    // Interleaved tile: per halo pixel {rcp(w^2+eps), img0, img1, img2}
    // -> one ds_load_b128 + one v_exp_f32 per tap in the hot loop.
    __shared__ __attribute__((aligned(16))) float4 s_t[NPIX];   // 24320 B

    const int t  = (int)threadIdx.x;
    const int bx = (int)blockIdx.x;
    const int by = (int)blockIdx.y;
    const int gx0 = bx * TILE - 4;     // halo col origin (16B aligned vs row start)
    const int gy0 = by * TILE - KRAD;  // halo row origin

    // Uniform params -> scalar loads; hoisted out of everything.
    const float sx  = gp[0];
    const float sy  = gp[1];
    const float rho = gp[2];
    const float dd  = 2.0f * (1.0f - rho * rho);
    const float ca  = 1.0f / (dd * sx * sx);
    const float cb  = -2.0f * rho / (dd * sx * sy);
    const float cc  = 1.0f / (dd * sy * sy);

    // 1) Zero-prefill only where the halo can leave the image (block-uniform branch,
    //    so the extra barrier is uniform across the workgroup).
    const bool border = (bx == 0) | (by == 0) |
                        (bx == (int)gridDim.x - 1) | (by == (int)gridDim.y - 1);
    if (border) {
        for (int idx = t; idx < NPIX; idx += 256)
            s_t[idx] = make_float4(0.f, 0.f, 0.f, 0.f);
        __syncthreads();   // ds stores visible before async writes to same tile
    }

    // 2) Async-copy in-bounds halo pixels straight into LDS (per-lane EXEC predicated).
    for (int idx = t; idx < NPIX; idx += 256) {
        const int r  = idx / TPITCH;
        const int cl = idx - r * TPITCH;
        const int gy = gy0 + r;
        const int gx = gx0 + cl;
        if ((unsigned)gy < IMG_H && (unsigned)gx < IMG_W) {
            const long go = (long)gy * IMG_W + gx;
            float* l = (float*)&s_t[idx];
            cp32_g2l(wts + go,          l + 0);
            cp32_g2l(img + go,          l + 1);
            cp32_g2l(img + HW + go,     l + 2);
            cp32_g2l(img + 2 * HW + go, l + 3);
        }
    }
    __builtin_amdgcn_s_wait_asynccnt(0);   // drain this wave's ASYNCcnt
    __syncthreads();

    // 3) Hoist the per-neighbor reciprocal out of the 49-tap loop:
    //    s_t[.].x <- rcp(w^2 + eps). (OOB pixels: w=0 -> 1e5 -> exp2 underflows to 0.)
    for (int idx = t; idx < NPIX; idx += 256) {
        const float w = s_t[idx].x;
        s_t[idx].x = fast_rcp(w * w + EPS);
    }
    __syncthreads();

    // 4) Stencil: each thread computes 4 output pixels (rows ly0+{0,8,16,24}, col lx).
    const int lx  = t & 31;
    const int ly0 = t >> 5;

    float acc0[4] = {0.f, 0.f, 0.f, 0.f};
    float acc1[4] = {0.f, 0.f, 0.f, 0.f};
    float acc2[4] = {0.f, 0.f, 0.f, 0.f};
    float wsum[4] = {0.f, 0.f, 0.f, 0.f};

#pragma unroll 1
    for (int di = 0; di < 7; ++di) {
#pragma unroll
        for (int dj = 0; dj < 7; ++dj) {
            const float xv = (float)(KRAD - di);
            const float yv = (float)(KRAD - dj);
            // quad * log2(e): v = exp(-quad/(w^2+eps)) = exp2(-q * inv)
            const float q = (ca * xv * xv + cb * xv * yv + cc * yv * yv) * LOG2E;
            const int ccl = lx + dj + 1;           // halo col (origin at -4)
#pragma unroll
            for (int s = 0; s < 4; ++s) {
                const int rr = ly0 + 8 * s + di;   // halo row
                const float4 px = s_t[rr * TPITCH + ccl];   // ds_load_b128
                const float v  = fast_exp2(-q * px.x);      // single trans op per tap
                wsum[s] += v;
                acc0[s] = fmaf(px.y, v, acc0[s]);
                acc1[s] = fmaf(px.z, v, acc1[s]);
                acc2[s] = fmaf(px.w, v, acc2[s]);
            }
        }
    }

    // 5) Normalize + store (coalesced b32 per plane).
    const int gx = bx * TILE + lx;
#pragma unroll
    for (int s = 0; s < 4; ++s) {
        const int gy = by * TILE + ly0 + 8 * s;
        const float inv = fast_rcp(wsum[s] + EPS);
        const long o = (long)gy * IMG_W + gx;
        out[o]          = acc0[s] * inv;
        out[o + HW]     = acc1[s] * inv;
        out[o + 2 * HW] = acc2[s] * inv;
    }
}

extern "C" void kernel_launch(void* const* d_in, const int* in_sizes, int n_in,
                              void* d_out, int out_size, void* d_ws, size_t ws_size,
                              hipStream_t stream) {
    (void)in_sizes; (void)n_in; (void)d_ws; (void)ws_size; (void)out_size;
    const float* img = (const float*)d_in[0];
    const float* wts = (const float*)d_in[1];
    const float* gp  = (const float*)d_in[2];
    float* out = (float*)d_out;
    dim3 grid(IMG_W / TILE, IMG_H / TILE, 1);
    gauss_psf_kernel<<<grid, 256, 0, stream>>>(img, wts, gp, out);
}